// TopKPool_57629871177976
// MI455X (gfx1250) — compile-verified
//
#include <hip/hip_runtime.h>
#include <hip/hip_bf16.h>
#include <math.h>

// Problem constants from the reference
#define NN    8192
#define FF    512
#define KKEEP 4096

typedef __attribute__((ext_vector_type(2))) float v2f;
typedef __attribute__((ext_vector_type(8))) float v8f;

// ---------------------------------------------------------------------------
// Kernel 0: normInv = 1/||kernel||  (512 elements, one block)
// ---------------------------------------------------------------------------
__global__ __launch_bounds__(256) void norm_kernel(const float* __restrict__ kern,
                                                   float* __restrict__ normInv) {
    __shared__ float red[256];
    float s = 0.f;
    for (int i = threadIdx.x; i < FF; i += 256) { float v = kern[i]; s += v * v; }
    red[threadIdx.x] = s;
    __syncthreads();
    for (int off = 128; off > 0; off >>= 1) {
        if ((int)threadIdx.x < off) red[threadIdx.x] += red[threadIdx.x + off];
        __syncthreads();
    }
    if (threadIdx.x == 0) normInv[0] = rsqrtf(red[0]);
}

// ---------------------------------------------------------------------------
// Kernel 1: y = (X @ kernel) * normInv  via V_WMMA_F32_16X16X4_F32.
// One wave (32 threads) per 16 rows. B-matrix = kernel chunk replicated
// across all 16 N columns, so every column of D equals y for the 16 rows.
// A layout (16x4 f32): lanes 0-15 M=0..15 K={k0,k0+1}; lanes 16-31 K={k0+2,k0+3}.
// D layout (16x16 f32): lane L<16 VGPR v -> M=v; lane L>=16 -> M=v+8; N=L%16.
// ---------------------------------------------------------------------------
__global__ __launch_bounds__(32) void matvec_wmma(const float* __restrict__ X,
                                                  const float* __restrict__ kern,
                                                  const float* __restrict__ normInv,
                                                  float* __restrict__ y) {
    const int lane    = threadIdx.x;       // 0..31, wave32
    const int half    = lane >> 4;         // 0: K lo pair, 1: K hi pair
    const int m       = lane & 15;         // row within tile / N column
    const int rowBase = blockIdx.x * 16;
    const size_t row  = (size_t)(rowBase + m);

    v8f c = {};
    for (int k0 = 0; k0 < FF; k0 += 4) {
        const int kA = k0 + half * 2;
        v2f a, b;
        a.x = X[row * FF + kA + 0];
        a.y = X[row * FF + kA + 1];
        b.x = kern[kA + 0];                // replicated across N -> same per lane group
        b.y = kern[kA + 1];
        // (neg_a, A, neg_b, B, c_mod, C, reuse_a, reuse_b)
        c = __builtin_amdgcn_wmma_f32_16x16x4_f32(false, a, false, b,
                                                  (short)0, c, false, false);
    }
    const float ninv = normInv[0];
    if (m == 0) {                          // lane 0 -> rows 0..7, lane 16 -> rows 8..15
#pragma unroll
        for (int v = 0; v < 8; ++v)
            y[rowBase + half * 8 + v] = c[v] * ninv;
    }
}

// ---------------------------------------------------------------------------
// Kernel 2: single-workgroup top-K.
//  Phase 1: bitonic sort of 8192 (value,index) pairs in LDS, descending by
//           value, tie-break ascending index (matches jax.lax.top_k).
//  Phase 2: bitonic sort of the first 4096 indices, ascending.
// LDS: 8192*(4+4) = 64 KB  (WGP has 320 KB).
// ---------------------------------------------------------------------------
__global__ __launch_bounds__(1024) void topk_sort(const float* __restrict__ y,
                                                  int* __restrict__ idxOut) {
    __shared__ float val[NN];
    __shared__ int   key[NN];
    const int tid = threadIdx.x;

    for (int i = tid; i < NN; i += 1024) { val[i] = y[i]; key[i] = i; }
    __syncthreads();

    // Phase 1: sort all 8192 pairs, "ascending" order == descending value
    for (int k = 2; k <= NN; k <<= 1) {
        for (int j = k >> 1; j > 0; j >>= 1) {
            for (int i = tid; i < NN; i += 1024) {
                const int ixj = i ^ j;
                if (ixj > i) {
                    const bool  up = ((i & k) == 0);
                    const float vi = val[i], vj = val[ixj];
                    const int   ki = key[i], kj = key[ixj];
                    const bool inorder = (vi > vj) || (vi == vj && ki < kj);
                    if (up != inorder) {
                        val[i] = vj; val[ixj] = vi;
                        key[i] = kj; key[ixj] = ki;
                    }
                }
            }
            __syncthreads();
        }
    }

    // Phase 2: sort surviving indices key[0..KKEEP) ascending
    for (int k = 2; k <= KKEEP; k <<= 1) {
        for (int j = k >> 1; j > 0; j >>= 1) {
            for (int i = tid; i < KKEEP; i += 1024) {
                const int ixj = i ^ j;
                if (ixj > i) {
                    const bool up = ((i & k) == 0);
                    const int  ki = key[i], kj = key[ixj];
                    const bool inorder = ki < kj;
                    if (up != inorder) { key[i] = kj; key[ixj] = ki; }
                }
            }
            __syncthreads();
        }
    }

    for (int i = tid; i < KKEEP; i += 1024) idxOut[i] = key[i];
}

// ---------------------------------------------------------------------------
// Kernel 3: X_pooled[r] = X[idx[r]] * tanh(y[idx[r]]), float4 coalesced.
// ---------------------------------------------------------------------------
__global__ __launch_bounds__(128) void gather_x(const float* __restrict__ X,
                                                const float* __restrict__ y,
                                                const int* __restrict__ idx,
                                                float* __restrict__ out) {
    const int r = blockIdx.x;
    const int g = idx[r];
    const float gate = tanhf(y[g]);
    const float4* src = (const float4*)(X + (size_t)g * FF);
    float4*       dst = (float4*)(out + (size_t)r * FF);
    float4 v = src[threadIdx.x];
    v.x *= gate; v.y *= gate; v.z *= gate; v.w *= gate;
    dst[threadIdx.x] = v;
}

// ---------------------------------------------------------------------------
// Kernel 4: A_pooled[r][c] = A[idx[r]][idx[c]].
// Row of A = 32 KB, stays hot in L2 (192 MB); sorted ascending idx gives
// ~stride-2 locality in the column gather. Prefetch next gathered row.
// ---------------------------------------------------------------------------
__global__ __launch_bounds__(256) void gather_a(const float* __restrict__ A,
                                                const int* __restrict__ idx,
                                                float* __restrict__ out) {
    const int c  = blockIdx.x * 256 + threadIdx.x;
    const int r  = blockIdx.y;
    const int gr = idx[r];
    const int gc = idx[c];
    if (r + 1 < KKEEP) {
        // warm L2 for the next gathered row (global_prefetch_b8)
        __builtin_prefetch(A + (size_t)idx[r + 1] * NN + (size_t)gc, 0, 1);
    }
    out[(size_t)r * KKEEP + c] = A[(size_t)gr * NN + gc];
}

// ---------------------------------------------------------------------------
extern "C" void kernel_launch(void* const* d_in, const int* in_sizes, int n_in,
                              void* d_out, int out_size, void* d_ws, size_t ws_size,
                              hipStream_t stream) {
    const float* X    = (const float*)d_in[0];
    const float* A    = (const float*)d_in[1];
    const float* kern = (const float*)d_in[2];

    float* out = (float*)d_out;
    float* Xp  = out;                          // 4096 * 512
    float* Ap  = out + (size_t)KKEEP * FF;     // 4096 * 4096

    // workspace layout: [0] normInv | [256] y[8192] | [+32768] idx[4096]
    float* ws_norm = (float*)d_ws;
    float* ws_y    = (float*)((char*)d_ws + 256);
    int*   ws_idx  = (int*)((char*)d_ws + 256 + (size_t)NN * sizeof(float));

    norm_kernel<<<1, 256, 0, stream>>>(kern, ws_norm);
    matvec_wmma<<<NN / 16, 32, 0, stream>>>(X, kern, ws_norm, ws_y);
    topk_sort<<<1, 1024, 0, stream>>>(ws_y, ws_idx);
    gather_x<<<KKEEP, 128, 0, stream>>>(X, ws_y, ws_idx, Xp);
    dim3 gA(KKEEP / 256, KKEEP);
    gather_a<<<gA, 256, 0, stream>>>(A, ws_idx, Ap);
}